// NetVLADagg_37434934952664
// MI455X (gfx1250) — compile-verified
//
#include <hip/hip_runtime.h>

typedef __attribute__((ext_vector_type(2))) float v2f;
typedef __attribute__((ext_vector_type(8))) float v8f;

#define N_IMG 32
#define DCH   512
#define PPIX  3136      // 56*56
#define KCL   64
#define EPSF  1e-12f

// Workspace layout (float offsets). Total ~7.47M floats (~29.9 MB).
#define WS_ASUM  0                                  // N*K   sum_p a[n,k,p]
#define WS_GSSQ  (N_IMG*KCL)                        // N     global sum-of-squares
#define WS_APR   (WS_GSSQ + 32)                     // N*K*P a'[n,k,p] = a*inv_norm
#define WS_VRAW  (WS_APR + (size_t)N_IMG*KCL*PPIX)  // N*K*D raw vlad

// CDNA5 async global->LDS copy (ASYNCcnt path); 16B per lane.
__device__ __forceinline__ void async_b128(const float* lds_dst, const float* gsrc) {
  unsigned lds_off = (unsigned)(size_t)lds_dst;         // low 32b of generic = LDS offset
  unsigned long long ga = (unsigned long long)(size_t)gsrc;
  asm volatile("global_load_async_to_lds_b128 %0, %1, off"
               :: "v"(lds_off), "v"(ga) : "memory");
}
__device__ __forceinline__ void wait_async_le3() {
  asm volatile("s_wait_asynccnt 0x3" ::: "memory");
}
__device__ __forceinline__ void wait_async_0() {
  asm volatile("s_wait_asynccnt 0x0" ::: "memory");
}

// ---------------------------------------------------------------- kernel 0
__global__ void k0_init(float* __restrict__ ws) {
  int i = blockIdx.x * 256 + threadIdx.x;
  if (i < N_IMG * KCL + 32) ws[WS_ASUM + i] = 0.0f;   // a_sum and gssq contiguous
}

// ---------------------------------------------------------------- kernel 2
// logits(64 x 16px) via V_WMMA_F32_16X16X4_F32 with the per-pixel L2-norm
// fused into the same x stream (lane L / L+16 jointly see all 512 channels of
// one pixel column), then softmax over K in LDS; emits a'[k,p] = a*inv_norm
// and partial a_sum. Block = 4 waves, one 16-cluster tile each, same 16 px.
__global__ void __launch_bounds__(128)
k2_assign(const float* __restrict__ x, const float* __restrict__ w,
          float* __restrict__ ws) {
  __shared__ float lg[KCL * 17];   // padded: logits -> probabilities
  __shared__ float invs[16];       // per-pixel 1/max(||x||,eps)
  const int n    = blockIdx.y;
  const int p0   = blockIdx.x * 16;
  const int t    = threadIdx.x;
  const int lane = t & 31, wv = t >> 5;
  const int m16  = lane & 15;              // A-row M / B-col N for this lane
  const int off  = (lane < 16) ? 0 : 2;    // K-offset held by this half-wave

  const float* wr = w + (wv * 16 + m16) * DCH;                 // A: conv_w row
  const float* xb = x + (size_t)n * DCH * PPIX + p0 + m16;     // B: x column

  v8f acc0 = {}; v8f acc1 = {};
  float nss = 0.f;                         // partial sum-of-squares (this lane's d's)
#pragma unroll 2
  for (int d0 = 0; d0 < DCH; d0 += 8) {
    {
      float2 av = *reinterpret_cast<const float2*>(wr + d0 + off);
      v2f a; a.x = av.x; a.y = av.y;
      v2f b; b.x = xb[(size_t)(d0 + off) * PPIX];
             b.y = xb[(size_t)(d0 + off + 1) * PPIX];
      nss += b.x * b.x + b.y * b.y;
      acc0 = __builtin_amdgcn_wmma_f32_16x16x4_f32(false, a, false, b,
                                                   (short)0, acc0, false, false);
    }
    {
      float2 av = *reinterpret_cast<const float2*>(wr + d0 + 4 + off);
      v2f a; a.x = av.x; a.y = av.y;
      v2f b; b.x = xb[(size_t)(d0 + 4 + off) * PPIX];
             b.y = xb[(size_t)(d0 + 4 + off + 1) * PPIX];
      nss += b.x * b.x + b.y * b.y;
      acc1 = __builtin_amdgcn_wmma_f32_16x16x4_f32(false, a, false, b,
                                                   (short)0, acc1, false, false);
    }
  }
  // combine the two half-wave partials -> full ||x[:,p]||^2 on every lane
  float ssq  = nss + __shfl_xor(nss, 16, 32);
  float invc = 1.0f / fmaxf(sqrtf(ssq), EPSF);

  v8f acc = acc0 + acc1;
  const int rbase = wv * 16 + ((lane < 16) ? 0 : 8);
#pragma unroll
  for (int v = 0; v < 8; ++v) lg[(rbase + v) * 17 + m16] = acc[v] * invc;
  if (t < 16) invs[t] = invc;              // wave0 lanes 0..15 own columns 0..15
  __syncthreads();

  if (t < 16) {                            // softmax over K=64 per pixel
    float m = -3.4e38f;
    for (int k = 0; k < KCL; ++k) m = fmaxf(m, lg[k * 17 + t]);
    float s = 0.f;
    for (int k = 0; k < KCL; ++k) { float e = __expf(lg[k * 17 + t] - m); lg[k * 17 + t] = e; s += e; }
    float r = 1.0f / s;
    for (int k = 0; k < KCL; ++k) lg[k * 17 + t] *= r;
  }
  __syncthreads();

  if (t < KCL) {                           // partial a_sum for this pixel tile
    float s = 0.f;
    for (int c = 0; c < 16; ++c) s += lg[t * 17 + c];
    atomicAdd(ws + WS_ASUM + n * KCL + t, s);
  }
  float* ap = ws + WS_APR + (size_t)n * KCL * PPIX;
#pragma unroll
  for (int i = 0; i < 8; ++i) {            // coalesced a' writes (16 px rows)
    int idx = i * 128 + t;
    int k = idx >> 4, c = idx & 15;
    ap[(size_t)k * PPIX + p0 + c] = lg[k * 17 + c] * invs[c];
  }
}

// ---------------------------------------------------------------- kernel 3
// agg[k,d] = sum_p a'[k,p] * x[d,p] via WMMA, with double-buffered
// GLOBAL_LOAD_ASYNC_TO_LDS_B128 staging (ASYNCcnt), fused
// vlad_raw = agg - a_sum*centroid. Block = 8 waves: 4 k-tiles x 2 d-tiles over
// a 32-wide d slab; p is the reduction dim, staged in chunks of 32.
// LDS pitch 36 floats: 16B-aligned B128 targets + conflict-free WMMA reads.
#define APITCH 36
__global__ void __launch_bounds__(256)
k3_agg(const float* __restrict__ x, const float* __restrict__ cent,
       float* __restrict__ ws) {
  __shared__ __align__(16) float a_s[2][KCL * APITCH];  // a'[64][32] padded
  __shared__ __align__(16) float x_s[2][32  * APITCH];  // x[32 d][32 p] padded
  const int n     = blockIdx.y;
  const int dbase = blockIdx.x * 32;
  const int t     = threadIdx.x;
  const int lane  = t & 31, wv = t >> 5;
  const int kt = wv & 3, dt2 = wv >> 2;
  const int m16 = lane & 15;
  const int off = (lane < 16) ? 0 : 2;

  const float* ap = ws + WS_APR + (size_t)n * KCL * PPIX;
  const float* xb = x + (size_t)n * DCH * PPIX;

  // per-thread staging coordinates (3 B128 async issues per thread per chunk)
  const int ra0 = t >> 3,         ca0 = (t & 7) * 4;          // a' rows 0..31
  const int ra1 = (t + 256) >> 3, ca1 = (t & 7) * 4;          // a' rows 32..63
  const int rx  = t >> 3,         cx  = (t & 7) * 4;          // x rows 0..31

  // prologue: stage chunk 0 into buffer 0
  async_b128(&a_s[0][ra0 * APITCH + ca0], ap + (size_t)ra0 * PPIX + ca0);
  async_b128(&a_s[0][ra1 * APITCH + ca1], ap + (size_t)ra1 * PPIX + ca1);
  async_b128(&x_s[0][rx  * APITCH + cx ], xb + (size_t)(dbase + rx) * PPIX + cx);

  v8f acc0 = {}; v8f acc1 = {};
  int buf = 0;
  for (int c = 0; c < PPIX / 32; ++c) {
    if (c < PPIX / 32 - 1) {               // prefetch next chunk into other buffer
      const int pn = (c + 1) * 32;
      float* an = a_s[buf ^ 1]; float* xn = x_s[buf ^ 1];
      async_b128(&an[ra0 * APITCH + ca0], ap + (size_t)ra0 * PPIX + pn + ca0);
      async_b128(&an[ra1 * APITCH + ca1], ap + (size_t)ra1 * PPIX + pn + ca1);
      async_b128(&xn[rx  * APITCH + cx ], xb + (size_t)(dbase + rx) * PPIX + pn + cx);
      wait_async_le3();                    // oldest 3 (chunk c) retired, newest 3 in flight
    } else {
      wait_async_0();
    }
    __syncthreads();                       // chunk c visible block-wide

    const float* arow = a_s[buf] + (kt * 16 + m16) * APITCH;   // A row: cluster k
    const float* xrow = x_s[buf] + (dt2 * 16 + m16) * APITCH;  // B col: channel d
#pragma unroll
    for (int s = 0; s < 8; s += 2) {
      {
        int pr = s * 4 + off;
        v2f a; a.x = arow[pr]; a.y = arow[pr + 1];
        v2f b; b.x = xrow[pr]; b.y = xrow[pr + 1];
        acc0 = __builtin_amdgcn_wmma_f32_16x16x4_f32(false, a, false, b,
                                                     (short)0, acc0, false, false);
      }
      {
        int pr = (s + 1) * 4 + off;
        v2f a; a.x = arow[pr]; a.y = arow[pr + 1];
        v2f b; b.x = xrow[pr]; b.y = xrow[pr + 1];
        acc1 = __builtin_amdgcn_wmma_f32_16x16x4_f32(false, a, false, b,
                                                     (short)0, acc1, false, false);
      }
    }
    __syncthreads();                       // all waves done reading buf before overwrite
    buf ^= 1;
  }
  v8f acc = acc0 + acc1;

  const int d = dbase + dt2 * 16 + m16;
  const float* asum = ws + WS_ASUM + n * KCL;
  float* vr = ws + WS_VRAW + (size_t)n * KCL * DCH;
#pragma unroll
  for (int v = 0; v < 8; ++v) {
    int k = kt * 16 + v + ((lane < 16) ? 0 : 8);
    vr[(size_t)k * DCH + d] = acc[v] - asum[k] * cent[k * DCH + d];
  }
}

// ---------------------------------------------------------------- kernel 4
// Intra-normalize each (n,k) row of 512, write to out, accumulate global ssq.
__global__ void __launch_bounds__(256)
k4_intra(float* __restrict__ ws, float* __restrict__ out) {
  __shared__ float red[256];
  __shared__ float sinv;
  const int n = blockIdx.y, k = blockIdx.x, t = threadIdx.x;
  const float* vr = ws + WS_VRAW + ((size_t)n * KCL + k) * DCH;
  float v0 = vr[t], v1 = vr[t + 256];
  red[t] = v0 * v0 + v1 * v1;
  __syncthreads();
  for (int sft = 128; sft > 0; sft >>= 1) {
    if (t < sft) red[t] += red[t + sft];
    __syncthreads();
  }
  if (t == 0) {
    float ss  = red[0];
    float inv = 1.0f / fmaxf(sqrtf(ss), EPSF);
    sinv = inv;
    atomicAdd(ws + WS_GSSQ + n, ss * inv * inv);
  }
  __syncthreads();
  float inv = sinv;
  float* o = out + (size_t)n * KCL * DCH + (size_t)k * DCH;
  o[t] = v0 * inv;
  o[t + 256] = v1 * inv;
}

// ---------------------------------------------------------------- kernel 5
__global__ void k5_global(const float* __restrict__ ws, float* __restrict__ out) {
  const int n = blockIdx.x, t = threadIdx.x;
  const float inv = 1.0f / fmaxf(sqrtf(ws[WS_GSSQ + n]), EPSF);
  float* o = out + (size_t)n * KCL * DCH;
  for (int i = t; i < KCL * DCH; i += 256) o[i] *= inv;
}

// ----------------------------------------------------------------
extern "C" void kernel_launch(void* const* d_in, const int* in_sizes, int n_in,
                              void* d_out, int out_size, void* d_ws, size_t ws_size,
                              hipStream_t stream) {
  (void)in_sizes; (void)n_in; (void)out_size; (void)ws_size;
  const float* x    = (const float*)d_in[0];
  const float* w    = (const float*)d_in[1];
  const float* cent = (const float*)d_in[2];
  float* out = (float*)d_out;
  float* ws  = (float*)d_ws;

  k0_init   <<<dim3(9),              dim3(256), 0, stream>>>(ws);
  k2_assign <<<dim3(PPIX/16, N_IMG), dim3(128), 0, stream>>>(x, w, ws);
  k3_agg    <<<dim3(DCH/32, N_IMG),  dim3(256), 0, stream>>>(x, cent, ws);
  k4_intra  <<<dim3(KCL, N_IMG),     dim3(256), 0, stream>>>(ws, out);
  k5_global <<<dim3(N_IMG),          dim3(256), 0, stream>>>(ws, out);
}